// RuleBasedEmbedding_38878043963937
// MI455X (gfx1250) — compile-verified
//
#include <hip/hip_runtime.h>

#define VOCAB     32000
#define EMBED     256
#define NRULES    100
#define NTOK      (4 * 2048)           // B*S = 8192
#define TILE_M    64                   // 4 x 16-row WMMA M-tiles per block
#define MTILES    (TILE_M / 16)        // 4
#define APAD      4                    // LDS row padding (floats) -> bank-conflict-free
#define AROW      (EMBED + APAD)       // 260
#define MAXTILES  (NTOK / TILE_M + NRULES)   // upper bound on total tiles = 228

typedef __attribute__((ext_vector_type(2))) float v2f;
typedef __attribute__((ext_vector_type(8))) float v8f;

// ---------------- bucketing kernels ----------------

__global__ void rb_init(int* cnt, int* cur) {
    int i = blockIdx.x * blockDim.x + threadIdx.x;
    if (i < NRULES) { cnt[i] = 0; cur[i] = 0; }
}

__global__ void rb_hist(const int* __restrict__ token_ids,
                        const int* __restrict__ token_rules,
                        int* __restrict__ cnt) {
    int i = blockIdx.x * blockDim.x + threadIdx.x;
    if (i < NTOK) {
        int r = token_rules[token_ids[i]];
        atomicAdd(&cnt[r], 1);
    }
}

__global__ void rb_scan(const int* __restrict__ cnt,
                        int* __restrict__ off,
                        int* __restrict__ tileOff) {
    if (threadIdx.x == 0 && blockIdx.x == 0) {
        int s = 0, t = 0;
        for (int r = 0; r < NRULES; ++r) {
            off[r] = s; tileOff[r] = t;
            s += cnt[r];
            t += (cnt[r] + TILE_M - 1) / TILE_M;
        }
        off[NRULES] = s; tileOff[NRULES] = t;
    }
}

__global__ void rb_scatter(const int* __restrict__ token_ids,
                           const int* __restrict__ token_rules,
                           const int* __restrict__ off,
                           int* __restrict__ cur,
                           int* __restrict__ bucket) {
    int i = blockIdx.x * blockDim.x + threadIdx.x;
    if (i < NTOK) {
        int r = token_rules[token_ids[i]];
        int p = atomicAdd(&cur[r], 1);
        bucket[off[r] + p] = i;           // store flat token position
    }
}

// ---------------- rule-grouped WMMA GEMM ----------------
// One block = one 64-token tile of one rule (four 16-row WMMA M-tiles).
// 8 waves; wave w computes output columns [32w, 32w+32) via two 16x16 N-tiles.
// Each K-step: 4 B loads shared by 8 WMMAs (4 M-tiles x 2 N-tiles).

__global__ __launch_bounds__(256)
void rb_gemm(const float* __restrict__ base_emb,      // [VOCAB][EMBED]
             const float* __restrict__ rule_xform,    // [NRULES][EMBED][EMBED]
             const int*   __restrict__ token_ids,     // [NTOK]
             const int*   __restrict__ cnt,
             const int*   __restrict__ off,
             const int*   __restrict__ tileOff,       // [NRULES+1]
             const int*   __restrict__ bucket,        // [NTOK]
             float*       __restrict__ out)           // [NTOK][EMBED]
{
    __shared__ float A[TILE_M][AROW];   // 64 x 260 f32, padded (~65 KB)
    __shared__ int   toks[TILE_M];

    const int bid   = blockIdx.x;
    const int total = tileOff[NRULES];
    if (bid >= total) return;           // uniform per block -> EXEC stays full below

    // find rule r with tileOff[r] <= bid < tileOff[r+1] (uniform scalar loop)
    int r = 0;
    #pragma unroll 1
    for (int i = 1; i < NRULES; ++i)
        if (tileOff[i] <= bid) r = i;

    const int tile   = bid - tileOff[r];
    const int nvalid = min(TILE_M, cnt[r] - tile * TILE_M);
    const int base   = off[r] + tile * TILE_M;

    const int tid = threadIdx.x;
    if (tid < TILE_M)
        toks[tid] = (tid < nvalid) ? bucket[base + tid] : -1;
    __syncthreads();

    // Stage A tile: 64 rows x 256 f32 as float4 (4096 vec4 over 256 threads)
    #pragma unroll
    for (int v = tid; v < TILE_M * (EMBED / 4); v += 256) {
        int row = v >> 6;               // / (EMBED/4)
        int c4  = v & 63;               // % (EMBED/4)
        float4 val = make_float4(0.f, 0.f, 0.f, 0.f);
        int tokpos = toks[row];
        if (tokpos >= 0) {
            const float4* src =
                (const float4*)(base_emb + (size_t)token_ids[tokpos] * EMBED);
            val = src[c4];
        }
        *(float4*)&A[row][c4 * 4] = val;  // row stride 1040B: 16B-aligned
    }
    __syncthreads();

    const int wave = tid >> 5;          // 0..7
    const int lane = tid & 31;
    const int lrow = lane & 15;         // A row within M-tile / B-D column
    const int hi   = lane >> 4;         // 0: K=k,k+1 ; 1: K=k+2,k+3
    const int n0   = wave * 32;

    const float* M = rule_xform + (size_t)r * EMBED * EMBED;

    v8f acc[MTILES][2] = {};            // [M-tile][N-tile], 64 VGPRs

    #pragma unroll 2
    for (int k = 0; k < EMBED; k += 4) {
        // A fragments (16x4 f32): VGPR0 = {K | K+2}, VGPR1 = {K+1 | K+3}
        v2f a[MTILES];
        #pragma unroll
        for (int mt = 0; mt < MTILES; ++mt) {
            a[mt].x = A[16 * mt + lrow][k + 2 * hi + 0];
            a[mt].y = A[16 * mt + lrow][k + 2 * hi + 1];
        }

        // B fragments (4x16 f32), coalesced 64B per half-wave, L2-resident;
        // shared by all 4 M-tiles -> 8 WMMAs per 4 loads.
        const float* Mk = M + (size_t)(k + 2 * hi) * EMBED;
        v2f b[2];
        b[0].x = Mk[n0 + lrow];
        b[0].y = Mk[EMBED + n0 + lrow];
        b[1].x = Mk[n0 + 16 + lrow];
        b[1].y = Mk[EMBED + n0 + 16 + lrow];

        #pragma unroll
        for (int mt = 0; mt < MTILES; ++mt) {
            acc[mt][0] = __builtin_amdgcn_wmma_f32_16x16x4_f32(
                false, a[mt], false, b[0], (short)0, acc[mt][0], false, false);
            acc[mt][1] = __builtin_amdgcn_wmma_f32_16x16x4_f32(
                false, a[mt], false, b[1], (short)0, acc[mt][1], false, false);
        }
    }

    // D layout: VGPR v -> row (v + 8*hi), col = lrow (per 16x16 f32 C/D map)
    #pragma unroll
    for (int mt = 0; mt < MTILES; ++mt) {
        #pragma unroll
        for (int v = 0; v < 8; ++v) {
            int m = 16 * mt + v + 8 * hi;
            int tokpos = toks[m];
            if (tokpos >= 0) {
                float* dst = out + (size_t)tokpos * EMBED;
                dst[n0 + lrow]      = acc[mt][0][v];
                dst[n0 + 16 + lrow] = acc[mt][1][v];
            }
        }
    }
}

// ---------------- launcher ----------------

extern "C" void kernel_launch(void* const* d_in, const int* in_sizes, int n_in,
                              void* d_out, int out_size, void* d_ws, size_t ws_size,
                              hipStream_t stream) {
    const int*   token_ids   = (const int*)  d_in[0];   // [4,2048] int32
    const float* base_emb    = (const float*)d_in[1];   // [32000,256] f32
    const float* rule_xform  = (const float*)d_in[2];   // [100,256,256] f32
    const int*   token_rules = (const int*)  d_in[3];   // [32000] int32
    float*       out         = (float*)d_out;           // [4,2048,256] f32

    int* ws      = (int*)d_ws;
    int* cnt     = ws;            // [128]
    int* off     = ws + 128;      // [128]
    int* tileOff = ws + 256;      // [128]
    int* cur     = ws + 384;      // [128]
    int* bucket  = ws + 512;      // [NTOK]

    rb_init<<<1, 128, 0, stream>>>(cnt, cur);
    rb_hist<<<(NTOK + 255) / 256, 256, 0, stream>>>(token_ids, token_rules, cnt);
    rb_scan<<<1, 32, 0, stream>>>(cnt, off, tileOff);
    rb_scatter<<<(NTOK + 255) / 256, 256, 0, stream>>>(token_ids, token_rules,
                                                       off, cur, bucket);
    rb_gemm<<<MAXTILES, 256, 0, stream>>>(base_emb, rule_xform, token_ids,
                                          cnt, off, tileOff, bucket, out);
}